// SupConLoss_seq_59090160058540
// MI455X (gfx1250) — compile-verified
//
#include <hip/hip_runtime.h>
#include <math.h>

typedef __attribute__((ext_vector_type(2))) float v2f;
typedef __attribute__((ext_vector_type(8))) float v8f;

#define B_N   4096
#define D_K   256
#define L_N   20
#define NCLS  5
#define N_OH  128              // 100 one-hot columns padded to 128
#define TEMP_INV (1.0f / 0.07f)
#define EPS_F 1e-20f

#define WMMA_F32(A, Bf, C) \
  __builtin_amdgcn_wmma_f32_16x16x4_f32(false, (A), false, (Bf), (short)0, (C), false, false)

// ---------------------------------------------------------------------------
// Kernel 1: logits = (F @ F^T) / T via f32 WMMA, register-blocked 16x64/wave.
// A-frag (16x4 f32): lane m=lane&15 -> row M; group g=lane>>4 holds K=2g,2g+1
//   -> one contiguous float2 per K-step, reused across 4 column tiles.
// B-frag mirrors A (B = A^T of the column tile) -> same float2 pattern.
// C/D layout: VGPR r -> (M = r + 8g, N = lane&15).
// ---------------------------------------------------------------------------
__global__ __launch_bounds__(256) void gemm_logits(const float* __restrict__ F,
                                                   float* __restrict__ logits) {
  const int tid  = threadIdx.x;
  const int wave = tid >> 5;
  const int lane = tid & 31;
  const int w    = blockIdx.x * 8 + wave;   // 256 rowTiles x 64 colGroups = 16384 waves
  const int tileRow  = w >> 6;
  const int colGroup = w & 63;
  const int rowBase  = tileRow << 4;
  const int colBase  = colGroup << 6;       // 64 columns per wave

  const int g = lane >> 4;
  const int m = lane & 15;

  const float* arow = F + (size_t)(rowBase + m) * D_K + 2 * g;
  const float* br0  = F + (size_t)(colBase +  0 + m) * D_K + 2 * g;
  const float* br1  = F + (size_t)(colBase + 16 + m) * D_K + 2 * g;
  const float* br2  = F + (size_t)(colBase + 32 + m) * D_K + 2 * g;
  const float* br3  = F + (size_t)(colBase + 48 + m) * D_K + 2 * g;

  v8f c0 = {}, c1 = {}, c2 = {}, c3 = {};
#pragma unroll 4
  for (int k0 = 0; k0 < D_K; k0 += 4) {
    v2f a  = *(const v2f*)(arow + k0);
    v2f q0 = *(const v2f*)(br0 + k0);
    v2f q1 = *(const v2f*)(br1 + k0);
    v2f q2 = *(const v2f*)(br2 + k0);
    v2f q3 = *(const v2f*)(br3 + k0);
    c0 = WMMA_F32(a, q0, c0);
    c1 = WMMA_F32(a, q1, c1);
    c2 = WMMA_F32(a, q2, c2);
    c3 = WMMA_F32(a, q3, c3);
  }

#pragma unroll
  for (int r = 0; r < 8; ++r) {
    const size_t rowOff = (size_t)(rowBase + r + 8 * g) * B_N + colBase + m;
    logits[rowOff +  0] = c0[r] * TEMP_INV;
    logits[rowOff + 16] = c1[r] * TEMP_INV;
    logits[rowOff + 32] = c2[r] * TEMP_INV;
    logits[rowOff + 48] = c3[r] * TEMP_INV;
  }
}

// ---------------------------------------------------------------------------
// Kernel 2: transpose labels (B x L) -> (L x B).
// ---------------------------------------------------------------------------
__global__ void transpose_labels(const int* __restrict__ labels,
                                 int* __restrict__ labT) {
  int idx = blockIdx.x * blockDim.x + threadIdx.x;
  if (idx < B_N * L_N) {
    int j = idx / L_N, l = idx % L_N;
    labT[l * B_N + j] = labels[j * L_N + l];
  }
}

// ---------------------------------------------------------------------------
// Kernel 3: class histogram per label column (deterministic, no atomics).
// ---------------------------------------------------------------------------
__global__ void class_counts(const int* __restrict__ labels,
                             int* __restrict__ counts) {
  int t = threadIdx.x;
  if (t < L_N * NCLS) {
    int l = t / NCLS, c = t % NCLS;
    int cnt = 0;
    for (int j = 0; j < B_N; ++j) cnt += (labels[j * L_N + l] == c) ? 1 : 0;
    counts[t] = cnt;
  }
}

// ---------------------------------------------------------------------------
// Kernel 4: transposed one-hot matrix, onehotT[(5l+c)][j] = (lab_l[j]==c).
// Rows 100..127 are zero padding. 128 x 4096 f32 = 2 MB.
// ---------------------------------------------------------------------------
__global__ void build_onehotT(const int* __restrict__ labT,
                              float* __restrict__ onehotT) {
  int idx = blockIdx.x * blockDim.x + threadIdx.x;   // 0 .. 128*4096-1
  int r = idx >> 12;
  int j = idx & (B_N - 1);
  float v = 0.0f;
  if (r < L_N * NCLS) {
    int l = r / NCLS, c = r % NCLS;
    v = (labT[l * B_N + j] == c) ? 1.0f : 0.0f;
  }
  onehotT[idx] = v;
}

// ---------------------------------------------------------------------------
// Kernel 5: G = logits @ onehot  (4096 x 128, K = 4096) via f32 WMMA.
// G[i][5l+c] = sum_j logits[i][j] * (lab_l[j]==c)  -> masked row sums.
// 256 row tiles x 8 col tiles = 2048 waves.
// ---------------------------------------------------------------------------
__global__ __launch_bounds__(256) void gemm_masked(const float* __restrict__ logits,
                                                   const float* __restrict__ onehotT,
                                                   float* __restrict__ G) {
  const int tid  = threadIdx.x;
  const int wave = tid >> 5;
  const int lane = tid & 31;
  const int w    = blockIdx.x * 8 + wave;
  const int tileRow = w >> 3;
  const int tileCol = w & 7;
  const int rowBase = tileRow << 4;
  const int colBase = tileCol << 4;

  const int g = lane >> 4;
  const int m = lane & 15;

  const float* arow = logits  + (size_t)(rowBase + m) * B_N + 2 * g;
  const float* brow = onehotT + (size_t)(colBase + m) * B_N + 2 * g;

  v8f c = {};
#pragma unroll 8
  for (int k0 = 0; k0 < B_N; k0 += 4) {
    v2f a = *(const v2f*)(arow + k0);
    v2f b = *(const v2f*)(brow + k0);
    c = WMMA_F32(a, b, c);
  }

  const int col = colBase + m;
#pragma unroll
  for (int r = 0; r < 8; ++r) {
    G[(size_t)(rowBase + r + 8 * g) * N_OH + col] = c[r];
  }
}

// ---------------------------------------------------------------------------
// Kernel 6: per-row max + exp-sum sweeps, then 20 per-label finals from G.
// ---------------------------------------------------------------------------
__global__ __launch_bounds__(256) void row_reduce(const float* __restrict__ logits,
                                                  const float* __restrict__ G,
                                                  const int* __restrict__ labT,
                                                  const int* __restrict__ counts,
                                                  float* __restrict__ rowLoss,
                                                  float* __restrict__ rowSingle) {
  __shared__ float red[256];
  __shared__ float mSh, eSh;

  const int i   = blockIdx.x;
  const int tid = threadIdx.x;
  const float* row = logits + (size_t)i * B_N;

  // ---- sweep 1: row max (diagonal included, as in reference) ----
  float mx = -INFINITY;
  for (int j = tid; j < B_N; j += 256) mx = fmaxf(mx, row[j]);
  red[tid] = mx;
  __syncthreads();
  for (int s = 128; s > 0; s >>= 1) {
    if (tid < s) red[tid] = fmaxf(red[tid], red[tid + s]);
    __syncthreads();
  }
  if (tid == 0) mSh = red[0];
  __syncthreads();
  const float m = mSh;

  // ---- sweep 2: exp-sum excluding diagonal ----
  float e = 0.0f;
  for (int j = tid; j < B_N; j += 256) {
    if (j != i) e += expf(row[j] - m);
  }
  __syncthreads();
  red[tid] = e;
  __syncthreads();
  for (int s = 128; s > 0; s >>= 1) {
    if (tid < s) red[tid] += red[tid + s];
    __syncthreads();
  }
  if (tid == 0) eSh = red[0];
  __syncthreads();

  // ---- per-label finals: one lane per label ----
  if (tid < L_N) {
    int   c      = labT[tid * B_N + i];
    float diag   = row[i];
    float S      = G[(size_t)i * N_OH + tid * NCLS + c] - diag;  // exclude self
    float pos    = (float)(counts[tid * NCLS + c] - 1);
    float single = (pos == 0.0f) ? 1.0f : 0.0f;
    float mean_lp = (S - pos * m) / (pos + single) - logf(eSh + EPS_F);
    rowLoss[tid * B_N + i]   = -mean_lp * (1.0f - single);
    rowSingle[tid * B_N + i] = single;
  }
}

// ---------------------------------------------------------------------------
// Kernel 7: deterministic finalize.
// ---------------------------------------------------------------------------
__global__ __launch_bounds__(256) void finalize(const float* __restrict__ rowLoss,
                                                const float* __restrict__ rowSingle,
                                                float* __restrict__ out) {
  __shared__ float red[256];
  __shared__ float total;
  int tid = threadIdx.x;
  if (tid == 0) total = 0.0f;
  __syncthreads();
  for (int l = 0; l < L_N; ++l) {
    float sL = 0.0f, sS = 0.0f;
    for (int j = tid; j < B_N; j += 256) {
      sL += rowLoss[l * B_N + j];
      sS += rowSingle[l * B_N + j];
    }
    red[tid] = sL;
    __syncthreads();
    for (int s = 128; s > 0; s >>= 1) { if (tid < s) red[tid] += red[tid + s]; __syncthreads(); }
    float SL = red[0];
    __syncthreads();
    red[tid] = sS;
    __syncthreads();
    for (int s = 128; s > 0; s >>= 1) { if (tid < s) red[tid] += red[tid + s]; __syncthreads(); }
    if (tid == 0) total += SL / ((float)B_N - red[0]);
    __syncthreads();
  }
  if (tid == 0) out[0] = total / (float)L_N;
}

// ---------------------------------------------------------------------------
extern "C" void kernel_launch(void* const* d_in, const int* in_sizes, int n_in,
                              void* d_out, int out_size, void* d_ws, size_t ws_size,
                              hipStream_t stream) {
  const float* F      = (const float*)d_in[0];   // (4096, 256) f32
  const int*   labels = (const int*)d_in[1];     // (4096, 20)  i32
  float*       out    = (float*)d_out;

  char* ws = (char*)d_ws;
  size_t off = 0;
  float* logits    = (float*)(ws + off); off += (size_t)B_N * B_N * sizeof(float); // 64 MB
  int*   labT      = (int*)(ws + off);   off += (size_t)L_N * B_N * sizeof(int);
  int*   counts    = (int*)(ws + off);   off += 512;
  float* onehotT   = (float*)(ws + off); off += (size_t)N_OH * B_N * sizeof(float); // 2 MB
  float* G         = (float*)(ws + off); off += (size_t)B_N * N_OH * sizeof(float); // 2 MB
  float* rowLoss   = (float*)(ws + off); off += (size_t)L_N * B_N * sizeof(float);
  float* rowSingle = (float*)(ws + off); off += (size_t)L_N * B_N * sizeof(float);

  // GEMM1: 16384 waves (16x64 strip each), 8 waves/block.
  gemm_logits<<<16384 / 8, 256, 0, stream>>>(F, logits);
  transpose_labels<<<(B_N * L_N + 255) / 256, 256, 0, stream>>>(labels, labT);
  class_counts<<<1, 128, 0, stream>>>(labels, counts);
  build_onehotT<<<(N_OH * B_N) / 256, 256, 0, stream>>>(labT, onehotT);
  // GEMM2: 2048 waves (16x16 tile each, K=4096), 8 waves/block.
  gemm_masked<<<2048 / 8, 256, 0, stream>>>(logits, onehotT, G);
  row_reduce<<<B_N, 256, 0, stream>>>(logits, G, labT, counts, rowLoss, rowSingle);
  finalize<<<1, 256, 0, stream>>>(rowLoss, rowSingle, out);
}